// LocNet2dNT_75196287419030
// MI455X (gfx1250) — compile-verified
//
#include <hip/hip_runtime.h>
#include <hip/hip_bf16.h>

typedef __attribute__((ext_vector_type(2))) float v2f;
typedef __attribute__((ext_vector_type(8))) float v8f;

// ---------------------------------------------------------------------------
// ws layout (floats):
//   xT : [64 pos][48 k][128 b]            = 393216
//   Y1 : [4 pos2][8192 k][128 b]          = 4194304
//   Y2 : [2048 k][128 b]                  = 262144
//   Y3 : [512 o][128 b]                   = 65536
// total ~18.75 MB
// ---------------------------------------------------------------------------

// Repack x (B=128, C=3, 32, 32) into per-position B-matrices xT[pos][c*16+i*4+j][b]
__global__ __launch_bounds__(256) void xpose_kernel(const float* __restrict__ x,
                                                    float* __restrict__ xT) {
  const int pos = blockIdx.x;          // 64 positions (p*8+q)
  const int p = pos >> 3, q = pos & 7;
  for (int idx = threadIdx.x; idx < 48 * 128; idx += 256) {
    const int k = idx >> 7;            // 0..47
    const int b = idx & 127;
    const int c = k >> 4;
    const int f = k & 15;
    const int i = f >> 2, j = f & 3;
    xT[(size_t)pos * 6144 + idx] =
        x[(((size_t)b * 3 + c) * 32 + (p * 4 + i)) * 32 + (q * 4 + j)];
  }
}

// LDS pair-interleaved layout: element (k, col) of a 16x128 chunk lives at
//   float index (k>>1)*288 + col*2 + (k&1)
// so the per-lane WMMA B operand {B[kk][n], B[kk+1][n]} is one aligned b64 load.
#define PAIR_ROW 288
#define CHUNK_FLOATS (8 * PAIR_ROW)    // 2304 floats per buffer

// Generic locally-connected GEMM layer:
//   D[o][b] = relu( scale * sum_k A[o*lda + (k>>4)*cs + (k&15) + pos*16] * B[pos*K*128 + k*128 + b] )
// Block = 256 thr = 8 waves; wave w computes rows [blockIdx.x*128 + w*16, +16) x all 128 cols.
__global__ __launch_bounds__(256) void locgemm_kernel(
    const float* __restrict__ A, const float* __restrict__ B,
    float* __restrict__ C, int K, int lda, int cs, int ostride, int cmode,
    float scale) {
  __shared__ float bs[2][CHUNK_FLOATS];   // double-buffered B chunks (18 KB)

  const int tid    = threadIdx.x;
  const int wave   = tid >> 5;
  const int lane   = tid & 31;
  const int lane16 = lane & 15;
  const int half   = lane >> 4;
  const int pos    = blockIdx.y;
  const int mbase  = blockIdx.x * 128;
  const int orow   = mbase + wave * 16 + lane16;

  const float* __restrict__ Bp   = B + (size_t)pos * (size_t)K * 128;
  const float* __restrict__ arow = A + (size_t)orow * (size_t)lda + (size_t)pos * 16;

  size_t cbase;
  if (cmode == 0) {                    // layer1 -> Y1[pos2][o*16+f2][b]
    const int p1 = pos >> 3, q1 = pos & 7;
    const int pos2 = (p1 >> 2) * 2 + (q1 >> 2);
    const int f2   = (p1 & 3) * 4 + (q1 & 3);
    cbase = (size_t)pos2 * (8192u * 128u) + (size_t)f2 * 128u;
  } else if (cmode == 1) {             // layer2 -> Y2[o*4+pos][b]
    cbase = (size_t)pos * 128u;
  } else {                             // layer3 -> Y3[o][b]
    cbase = 0;
  }

  // per-thread staging coordinates (8 async b32 copies per thread per chunk)
  const int skr  = tid >> 7;           // handles k rows {skr, skr+2, ...} pattern below
  (void)skr;

  v8f zero = {};
  v8f acc[8];
#pragma unroll
  for (int i = 0; i < 8; ++i) acc[i] = zero;

  const int nch = K >> 4;              // 16-k chunks

  // ---- async staging of one 16x128 chunk into buf (global -> LDS, no VGPR data)
  auto stage_async = [&](int c) {
    const float* gbase = Bp + (size_t)c * 2048;   // chunk is contiguous 8KB
#pragma unroll
    for (int t = 0; t < 8; ++t) {
      const int i   = tid + t * 256;   // 0..2047
      const int kr  = i >> 7;
      const int col = i & 127;
      const unsigned loff =
          (unsigned)(unsigned long long)&bs[c & 1]
                                          [(kr >> 1) * PAIR_ROW + (col << 1) + (kr & 1)];
      const unsigned goff = (unsigned)i << 2;
      asm volatile("global_load_async_to_lds_b32 %0, %1, %2"
                   :: "v"(loff), "v"(goff), "s"(gbase)
                   : "memory");
    }
  };

  stage_async(0);                      // prologue

  for (int c = 0; c < nch; ++c) {
    if (c + 1 < nch) {
      stage_async(c + 1);              // overlap next chunk's stream with compute
      asm volatile("s_wait_asynccnt 0x8" ::: "memory");  // retire chunk c only
    } else {
      asm volatile("s_wait_asynccnt 0x0" ::: "memory");
    }
    __syncthreads();

    const float* __restrict__ ac = arow + (size_t)c * (size_t)cs;
    if (c + 1 < nch)                   // prefetch next weight chunk for this row
      __builtin_prefetch(arow + (size_t)(c + 1) * (size_t)cs, 0, 3);

    const float* __restrict__ bbuf = &bs[c & 1][0];

#pragma unroll
    for (int k0 = 0; k0 < 16; k0 += 4) {
      const int kk = k0 + 2 * half;    // per-lane K pair: {kk, kk+1}
      v2f a = *(const v2f*)(ac + kk);
      const int prow = ((k0 >> 1) + half) * PAIR_ROW + (lane16 << 1);
#pragma unroll
      for (int nt = 0; nt < 8; ++nt) {
        v2f b = *(const v2f*)(bbuf + prow + nt * 32);
        acc[nt] = __builtin_amdgcn_wmma_f32_16x16x4_f32(
            false, a, false, b, (short)0, acc[nt], false, false);
      }
    }
    __syncthreads();                   // all reads done before buffer is rewritten
  }

  // scale + relu + scatter-store in next layer's B layout
#pragma unroll
  for (int nt = 0; nt < 8; ++nt) {
#pragma unroll
    for (int j = 0; j < 8; ++j) {
      float v = acc[nt][j] * scale;
      v = v > 0.f ? v : 0.f;
      const int row = mbase + wave * 16 + j + 8 * half;  // D vgpr j -> M = j + 8*half
      const int col = nt * 16 + lane16;
      C[cbase + (size_t)row * (size_t)ostride + col] = v;
    }
  }
}

// out[b] = (sum_o Y3[o][b] * beta[o]) / sqrt(512)
__global__ __launch_bounds__(128) void head_kernel(const float* __restrict__ Y3,
                                                   const float* __restrict__ beta,
                                                   float* __restrict__ out) {
  const int b = threadIdx.x;
  float acc = 0.f;
#pragma unroll 8
  for (int o = 0; o < 512; ++o) acc += Y3[(size_t)o * 128 + b] * beta[o];
  out[b] = acc * 0.04419417382415922f;  // 1/sqrt(512)
}

extern "C" void kernel_launch(void* const* d_in, const int* in_sizes, int n_in,
                              void* d_out, int out_size, void* d_ws, size_t ws_size,
                              hipStream_t stream) {
  const float* x    = (const float*)d_in[0];
  const float* w1   = (const float*)d_in[1];
  const float* w2   = (const float*)d_in[2];
  const float* w3   = (const float*)d_in[3];
  const float* beta = (const float*)d_in[4];

  float* ws = (float*)d_ws;
  float* xT = ws;                        // 393216
  float* Y1 = xT + 393216;               // 4194304
  float* Y2 = Y1 + 4194304;              // 262144
  float* Y3 = Y2 + 262144;               // 65536

  xpose_kernel<<<64, 256, 0, stream>>>(x, xT);

  // L1: 64 pos, M=512 K=48   lda=3072  cs=1024 ; out -> Y1  (ostride 16*128)
  locgemm_kernel<<<dim3(4, 64), 256, 0, stream>>>(
      w1, xT, Y1, 48, 3072, 1024, 2048, 0, 0.14433756729740643f);   // 1/sqrt(48)

  // L2: 4 pos, M=512 K=8192  lda=32768 cs=64   ; out -> Y2  (ostride 4*128)
  locgemm_kernel<<<dim3(4, 4), 256, 0, stream>>>(
      w2, Y1, Y2, 8192, 32768, 64, 512, 1, 0.011048543456039806f);  // 1/sqrt(8192)

  // L3: 1 pos, M=512 K=2048  lda=2048  cs=16   ; out -> Y3  (ostride 128)
  locgemm_kernel<<<dim3(4, 1), 256, 0, stream>>>(
      w3, Y2, Y3, 2048, 2048, 16, 128, 2, 0.022097086912079608f);   // 1/sqrt(2048)

  head_kernel<<<1, 128, 0, stream>>>(Y3, beta, (float*)d_out);
}